// AMFMSynth_9586367004686
// MI455X (gfx1250) — compile-verified
//
#include <hip/hip_runtime.h>
#include <hip/hip_bf16.h>
#include <math.h>

typedef float v2f __attribute__((ext_vector_type(2)));
typedef float v8f __attribute__((ext_vector_type(8)));

namespace {
constexpr int   kB      = 256;
constexpr int   kN      = 65536;
constexpr float kSR     = 44100.0f;
constexpr float kTwoPi  = 6.28318530717958647692f;
constexpr int   kChunk  = 1024;                 // samples per wave
constexpr int   kNChunk = kN / kChunk;          // 64 chunks per voice
constexpr int   kTiles  = kChunk / 256;         // 4 WMMA tiles per chunk
}

// Broadcast from a fixed lane: v_readlane_b32 (no LDS / ds_bpermute traffic).
__device__ __forceinline__ float readlane_f(float v, int l) {
    return __int_as_float(__builtin_amdgcn_readlane(__float_as_int(v), l));
}

// One wave = one (voice, 1024-sample chunk). Chunk-start phase comes from the
// closed-form Dirichlet sum; within the chunk the cumsum is done on the matrix
// pipe via triangular-matrix WMMAs (16x16x4 f32), tile total carried serially.
__global__ __launch_bounds__(256) void amfm_synth_kernel(
    const float* __restrict__ theta_am_0to1,
    const float* __restrict__ theta_fm_0to1,
    const float* __restrict__ phase,
    const float* __restrict__ phase_am,
    const float* __restrict__ phase_fm,
    const float* __restrict__ u_am_mi,
    const float* __restrict__ u_fm_hz,
    const float* __restrict__ u_f0_hz,
    float* __restrict__ out)
{
    const int lane  = threadIdx.x & 31;
    const int gwave = blockIdx.x * (blockDim.x >> 5) + (threadIdx.x >> 5);
    const int b     = gwave / kNChunk;
    const int chunk = gwave % kNChunk;
    const int s0    = chunk * kChunk;
    if (b >= kB) return;                    // wave-uniform, never taken

    const int  m  = lane & 15;              // row (A side) == column (B/C side)
    const bool hi = lane >= 16;

    // ---- per-voice parameters (wave-uniform scalar loads) ----
    const float am_hz = exp2f(theta_am_0to1[b] * 4.0f - 1.0f);               // [0.5,8]
    const float mi_am = u_am_mi[b];                                          // [0,1]
    const float mi_fm = theta_fm_0to1[b];                                    // [0,1]
    const float fm_hz = exp2f(u_fm_hz[b] * 4.0f - 1.0f);                     // [0.5,8]
    const float f0    = exp2f(u_f0_hz[b] * (log2f(523.25f) - log2f(32.7f))
                              + log2f(32.7f));                               // [32.7,523.25]

    const float c1   = kTwoPi * f0 / kSR;        // per-sample carrier phase step
    const float c2   = c1 * mi_fm;               // FM deviation coefficient
    const float dfm  = kTwoPi * fm_hz / kSR;
    const float afm  = kTwoPi * phase_fm[b];
    const float dam  = kTwoPi * am_hz / kSR;
    const float aam  = kTwoPi * phase_am[b];
    const float base = kTwoPi * phase[b];

    // Chunk-start carry: sum_{k=0}^{s0-1} (c1 + c2*sin(afm + k*dfm)), closed form.
    float carry = 0.0f;
    if (s0 > 0) {
        const float mm = (float)(s0 - 1);
        const float S  = __sinf(0.5f * (mm + 1.0f) * dfm) *
                         __sinf(afm + 0.5f * mm * dfm) / __sinf(0.5f * dfm);
        carry = c1 * (float)s0 + c2 * S;
    }

    // Loop-invariant triangular WMMA operands, computed once per wave:
    //   BmC[q] = U[k0:k0+4, :]      (U[k][c] = k<=c)  B(4x16) layout
    //   A2C[q] = Lstrict[:, k0:k0+4] (L[m][k] = k<m)   A(16x4) layout
    v2f BmC[4], A2C[4];
    #pragma unroll
    for (int q = 0; q < 4; ++q) {
        const int k0 = 4 * q;
        BmC[q].x = (m >= (k0 + (hi ? 1 : 0))) ? 1.0f : 0.0f;
        BmC[q].y = (m >= (k0 + (hi ? 3 : 2))) ? 1.0f : 0.0f;
        A2C[q].x = ((k0 + (hi ? 2 : 0)) < m) ? 1.0f : 0.0f;
        A2C[q].y = ((k0 + (hi ? 3 : 1)) < m) ? 1.0f : 0.0f;
    }

    // Per-lane affine bases (everything below is incremental FMAs + imm offsets).
    const int   laneRow0 = (hi ? 128 : 0) + m;            // 16*row(i=0) + m
    float* __restrict__ p0 = out + (size_t)b * kN + s0 + laneRow0;
    const float fn0    = (float)(s0 + laneRow0);          // sample idx at (t=0,i=0)
    const float dam16  = dam * 16.0f;
    const float dam256 = dam * 256.0f;
    const float dfm4   = dfm * 4.0f;
    const float vb0    = dfm * (float)(s0 + 16 * m + (hi ? 2 : 0)) + afm;
    const float dfm256 = dfm * 256.0f;

    #pragma unroll
    for (int t = 0; t < kTiles; ++t) {
        // v[j] = phase step of sample n0+j; lane holds A-matrix slots
        // A(16x4) layout: lanes0-15 -> K0,K1 ; lanes16-31 -> K2,K3 (row = lane%16)
        const float vbase = vb0 + dfm256 * (float)t;
        float vv[8];
        #pragma unroll
        for (int q = 0; q < 4; ++q) {
            const float aq = vbase + dfm4 * (float)q;
            vv[2*q]   = c1 + c2 * __sinf(aq);
            vv[2*q+1] = c1 + c2 * __sinf(aq + dfm);
        }

        // S1 = V x U: inclusive prefix inside each 16-sample segment
        v8f s1 = {};
        #pragma unroll
        for (int q = 0; q < 4; ++q) {
            v2f A;  A.x = vv[2*q];  A.y = vv[2*q+1];
            s1 = __builtin_amdgcn_wmma_f32_16x16x4_f32(false, A, false, BmC[q],
                                                       (short)0, s1, false, false);
        }

        // D2 = Lstrict x broadcast(rowTotals) + S1 -> full 256-sample inclusive cumsum
        // rowTotals R[r] = S1[r][15]: readlane 15 (r<8) / 31 (r>=8) of D-VGPR r%8
        v8f d2 = s1;
        #pragma unroll
        for (int q = 0; q < 4; ++q) {
            const int k0 = 4 * q;
            const float R0 = readlane_f(s1[(k0+0)&7], (k0+0) < 8 ? 15 : 31);
            const float R1 = readlane_f(s1[(k0+1)&7], (k0+1) < 8 ? 15 : 31);
            const float R2 = readlane_f(s1[(k0+2)&7], (k0+2) < 8 ? 15 : 31);
            const float R3 = readlane_f(s1[(k0+3)&7], (k0+3) < 8 ? 15 : 31);
            v2f BR; BR.x = hi ? R1 : R0;  BR.y = hi ? R3 : R2;
            d2 = __builtin_amdgcn_wmma_f32_16x16x4_f32(false, A2C[q], false, BR,
                                                       (short)0, d2, false, false);
        }
        const float tileTotal = readlane_f(d2[7], 31);   // D2[15][15] = sum of tile

        // synthesize + streaming NT stores (D 8-VGPR layout: VGPR i -> rows i, i+8)
        const float bc  = base + carry;                  // one add per tile
        const float amT = dam * fn0 + aam + dam256 * (float)t;
        float* __restrict__ pt = p0 + t * 256;
        #pragma unroll
        for (int i = 0; i < 8; ++i) {
            const float arg = bc + d2[i];
            const float am  = mi_am * __sinf(amT + dam16 * (float)i);
            const float x   = 0.5f * __sinf(arg) * (1.0f + am);
            __builtin_nontemporal_store(x, pt + 16 * i);
        }
        carry += tileTotal;
    }
}

extern "C" void kernel_launch(void* const* d_in, const int* in_sizes, int n_in,
                              void* d_out, int out_size, void* d_ws, size_t ws_size,
                              hipStream_t stream) {
    (void)in_sizes; (void)n_in; (void)d_ws; (void)ws_size; (void)out_size;
    const float* theta_am = (const float*)d_in[0];
    const float* theta_fm = (const float*)d_in[1];
    const float* phase    = (const float*)d_in[2];
    const float* phase_am = (const float*)d_in[3];
    const float* phase_fm = (const float*)d_in[4];
    const float* u_am_mi  = (const float*)d_in[5];
    const float* u_fm_hz  = (const float*)d_in[6];
    const float* u_f0_hz  = (const float*)d_in[7];
    float* out = (float*)d_out;

    constexpr int totalWaves = kB * kNChunk;            // 16384 waves
    dim3 block(256);                                    // 8 wave32 per block
    dim3 grid(totalWaves / (256 / 32));                 // 2048 blocks
    hipLaunchKernelGGL(amfm_synth_kernel, grid, block, 0, stream,
                       theta_am, theta_fm, phase, phase_am, phase_fm,
                       u_am_mi, u_fm_hz, u_f0_hz, out);
}